// graphcl_79637283602625
// MI455X (gfx1250) — compile-verified
//
#include <hip/hip_runtime.h>
#include <math.h>

#define B_   4096
#define D_   256
#define P_   100
#define K_   10
#define N_   8192
#define INVT 5.0f   // 1 / TEMPERATURE

typedef float v2f __attribute__((ext_vector_type(2)));
typedef float v8f __attribute__((ext_vector_type(8)));

// ---------------- row L2-normalize (block = 256 = D) ----------------
__global__ void k_norm(const float* __restrict__ in, float* __restrict__ out) {
    __shared__ float red[256];
    int row = blockIdx.x;
    int t   = threadIdx.x;
    float v = in[(size_t)row * D_ + t];
    red[t] = v * v;
    __syncthreads();
    for (int s = 128; s > 0; s >>= 1) {
        if (t < s) red[t] += red[t + s];
        __syncthreads();
    }
    float nrm = sqrtf(red[0]);
    float sc  = 1.0f / fmaxf(nrm, 1e-12f);
    out[(size_t)row * D_ + t] = v * sc;
}

// ---------------- prototype distance matrix pd = 1 - w w^T ----------------
__global__ void k_pd(const float* __restrict__ w, float* __restrict__ pd) {
    __shared__ float wa[D_];
    int a = blockIdx.x;
    int t = threadIdx.x; // 128 threads
    wa[t]       = w[(size_t)a * D_ + t];
    wa[t + 128] = w[(size_t)a * D_ + t + 128];
    __syncthreads();
    if (t < P_) {
        const float* wb = w + (size_t)t * D_;
        float acc = 0.f;
        for (int k = 0; k < D_; k++) acc += wa[k] * wb[k];
        pd[a * P_ + t] = 1.0f - acc;
    }
}

__global__ void k_zero_cnt(int* __restrict__ cnt) {
    if (threadIdx.x < P_) cnt[threadIdx.x] = 0;
}

// ---------------- prototype histogram + per-row membership bitmask ----------------
__global__ void k_count_mask(const int* __restrict__ hq, const int* __restrict__ nq,
                             int* __restrict__ cnt, uint4* __restrict__ negm) {
    int i = blockIdx.x * blockDim.x + threadIdx.x;
    if (i >= N_) return;
    atomicAdd(&cnt[hq[i]], 1);
    unsigned m0 = 0, m1 = 0, m2 = 0, m3 = 0;
    for (int k = 0; k < K_; k++) {
        int p = nq[i * K_ + k];
        unsigned bit = 1u << (p & 31);
        int w = p >> 5;
        if (w == 0) m0 |= bit;
        else if (w == 1) m1 |= bit;
        else if (w == 2) m2 |= bit;
        else m3 |= bit;
    }
    negm[i] = make_uint4(m0, m1, m2, m3);
}

// ---------------- per-row stats of the (virtual) rw matrix ----------------
// Row i of raw rw: value pd[hq[i],hq[j]] for j!=i, and 0 at j==i.
// Collapse over columns via cnt[p]; diagonal replaces one occurrence of hq[i] by 0.
__global__ void k_rowstats(const int* __restrict__ hq, const int* __restrict__ cnt,
                           const float* __restrict__ pd, float4* __restrict__ rs) {
    int i = blockIdx.x * blockDim.x + threadIdx.x;
    if (i >= N_) return;
    int hqi = hq[i];
    const float* row = pd + hqi * P_;
    float vmin = 0.f, vmax = 0.f;      // diagonal contributes exact 0
    float sv = 0.f, sv2 = 0.f;
    for (int p = 0; p < P_; p++) {
        int c = cnt[p] - (p == hqi ? 1 : 0);
        if (c <= 0) continue;
        float v = row[p];
        vmin = fminf(vmin, v);
        vmax = fmaxf(vmax, v);
        float fc = (float)c;
        sv  += fc * v;
        sv2 += fc * v * v;
    }
    float inv = 1.0f / (vmax - vmin);
    float n   = (float)N_;
    // normalized value: (v - vmin)*inv ; sums over all n columns (diag v=0 included in sv/sv2 as 0)
    float mu  = inv * (sv - n * vmin) / n;
    float sn2 = inv * inv * (sv2 - 2.f * vmin * sv + n * vmin * vmin);
    float var = (sn2 - n * mu * mu) / (n - 1.0f);   // ddof=1
    rs[i] = make_float4(vmin, inv, mu, 1.0f / (2.0f * var));
}

// ---------------- positive-pair dots ----------------
__global__ void k_pos(const float* __restrict__ outn, float* __restrict__ pos_s) {
    __shared__ float red[256];
    int i = blockIdx.x;
    int t = threadIdx.x;
    red[t] = outn[(size_t)i * D_ + t] * outn[(size_t)(i + B_) * D_ + t];
    __syncthreads();
    for (int s = 128; s > 0; s >>= 1) {
        if (t < s) red[t] += red[t + s];
        __syncthreads();
    }
    if (t == 0) pos_s[i] = red[0];
}

// ---------------- fused epilogue for one 16x16 WMMA fragment ----------------
__device__ __forceinline__ float tile_sum(
    const v8f& c, int ibase, int jbase, int l15, int hi,
    const float* __restrict__ pd, const int* __restrict__ hq,
    const float4* __restrict__ rs, const uint4* __restrict__ negm) {
    int j = jbase + l15;
    int hqj = hq[j];
    float4 rsj = rs[j];                 // .z = mu_j, .w = 1/(2 var_j)  (column broadcast quirk)
    unsigned shamt = (unsigned)(hqj & 31);
    int wsel = hqj >> 5;
    float acc = 0.f;
#pragma unroll
    for (int v = 0; v < 8; v++) {
        int i = ibase + hi * 8 + v;     // C/D layout: lanes 16-31 hold M+8
        int hqi = hq[i];
        float4 rsi = rs[i];             // .x = rmin_i, .y = 1/range_i
        uint4 m = negm[i];
        unsigned mw = (wsel == 0) ? m.x : (wsel == 1) ? m.y : (wsel == 2) ? m.z : m.w;
        bool ok = (i != j) && ((mw >> shamt) & 1u);
        float raw = pd[hqi * P_ + hqj];
        float t = (raw - rsi.x) * rsi.y - rsj.z;
        float arg = c[v] * INVT + t * t * rsj.w;
        acc += ok ? expf(arg) : 0.f;
    }
    return acc;
}

// ---------------- heavy kernel: G = out @ out^T via f32 WMMA, fused masked exp-sum ----------------
__global__ void __launch_bounds__(256) k_main(
    const float* __restrict__ outn, const float* __restrict__ pd,
    const int* __restrict__ hq, const float4* __restrict__ rs,
    const uint4* __restrict__ negm, float* __restrict__ partials) {
    int lane = threadIdx.x & 31;
    int wave = threadIdx.x >> 5;
    int tile = blockIdx.x * 8 + wave;    // 65536 tiles of 32x32
    int ti = tile >> 8;
    int tj = tile & 255;
    int i0 = ti * 32, j0 = tj * 32;
    int l15 = lane & 15;
    int hi  = lane >> 4;

    // f32 WMMA A(16x4)/B(4x16) fragment: lane<16 -> K={0,1}, lane>=16 -> K={2,3}
    const float* pA0 = outn + (size_t)(i0 + l15) * D_ + 2 * hi;
    const float* pA1 = pA0 + (size_t)16 * D_;
    const float* pB0 = outn + (size_t)(j0 + l15) * D_ + 2 * hi;
    const float* pB1 = pB0 + (size_t)16 * D_;

    v8f c00 = {0.f,0.f,0.f,0.f,0.f,0.f,0.f,0.f};
    v8f c01 = c00, c10 = c00, c11 = c00;
#pragma unroll 4
    for (int s = 0; s < 64; s++) {
        v2f a0 = *(const v2f*)(pA0 + 4 * s);
        v2f a1 = *(const v2f*)(pA1 + 4 * s);
        v2f b0 = *(const v2f*)(pB0 + 4 * s);
        v2f b1 = *(const v2f*)(pB1 + 4 * s);
        c00 = __builtin_amdgcn_wmma_f32_16x16x4_f32(false, a0, false, b0, (short)0, c00, false, false);
        c01 = __builtin_amdgcn_wmma_f32_16x16x4_f32(false, a0, false, b1, (short)0, c01, false, false);
        c10 = __builtin_amdgcn_wmma_f32_16x16x4_f32(false, a1, false, b0, (short)0, c10, false, false);
        c11 = __builtin_amdgcn_wmma_f32_16x16x4_f32(false, a1, false, b1, (short)0, c11, false, false);
    }

    float acc = 0.f;
    acc += tile_sum(c00, i0,      j0,      l15, hi, pd, hq, rs, negm);
    acc += tile_sum(c01, i0,      j0 + 16, l15, hi, pd, hq, rs, negm);
    acc += tile_sum(c10, i0 + 16, j0,      l15, hi, pd, hq, rs, negm);
    acc += tile_sum(c11, i0 + 16, j0 + 16, l15, hi, pd, hq, rs, negm);

    // deterministic wave + block reduction
    for (int off = 16; off > 0; off >>= 1) acc += __shfl_xor(acc, off, 32);
    __shared__ float wsum[8];
    if (lane == 0) wsum[wave] = acc;
    __syncthreads();
    if (threadIdx.x == 0) {
        float s = 0.f;
        for (int w = 0; w < 8; w++) s += wsum[w];
        partials[blockIdx.x] = s;
    }
}

// ---------------- final deterministic reduction + loss ----------------
__global__ void k_final(const float* __restrict__ partials, const float* __restrict__ pos_s,
                        float* __restrict__ outp) {
    __shared__ float red[256];
    int t = threadIdx.x;
    float s = 0.f;
    for (int i = t; i < 8192; i += 256) s += partials[i];
    red[t] = s;
    __syncthreads();
    for (int k = 128; k > 0; k >>= 1) {
        if (t < k) red[t] += red[t + k];
        __syncthreads();
    }
    float denom = red[0];
    __syncthreads();
    float p = 0.f;
    for (int i = t; i < B_; i += 256) p += pos_s[i];
    red[t] = p;
    __syncthreads();
    for (int k = 128; k > 0; k >>= 1) {
        if (t < k) red[t] += red[t + k];
        __syncthreads();
    }
    if (t == 0) {
        // loss = mean(-log(pos/denom)) = log(denom) - (sum_i s_i) / (B * T)
        outp[0] = logf(denom) - red[0] * (INVT / (float)B_);
    }
}

extern "C" void kernel_launch(void* const* d_in, const int* in_sizes, int n_in,
                              void* d_out, int out_size, void* d_ws, size_t ws_size,
                              hipStream_t stream) {
    const float* x  = (const float*)d_in[0];
    const float* xa = (const float*)d_in[1];
    const float* pw = (const float*)d_in[2];
    const int*   hq = (const int*)d_in[3];   // (2,B) flat == concat of both views
    const int*   nq = (const int*)d_in[4];   // (2,B,K) flat == (2B,K)

    char* ws = (char*)d_ws;
    float*  out_n = (float*)(ws);                         // 8192*256 f32  = 8388608 B
    float*  pd    = (float*)(ws + 8388608);               // 100*100 f32   = 40000 B
    float*  wn    = (float*)(ws + 8428608);               // 100*256 f32   = 102400 B
    int*    cnt   = (int*)  (ws + 8531008);               // 100 i32 (pad 512)
    uint4*  negm  = (uint4*)(ws + 8531520);               // 8192*16 B     = 131072 B
    float4* rs    = (float4*)(ws + 8662592);              // 8192*16 B     = 131072 B
    float*  pos_s = (float*)(ws + 8793664);               // 4096 f32      = 16384 B
    float*  part  = (float*)(ws + 8810048);               // 8192 f32      = 32768 B

    k_norm<<<B_,  256, 0, stream>>>(x,  out_n);
    k_norm<<<B_,  256, 0, stream>>>(xa, out_n + (size_t)B_ * D_);
    k_norm<<<P_,  256, 0, stream>>>(pw, wn);
    k_pd  <<<P_,  128, 0, stream>>>(wn, pd);
    k_zero_cnt<<<1, 128, 0, stream>>>(cnt);
    k_count_mask<<<32, 256, 0, stream>>>(hq, nq, cnt, negm);
    k_rowstats  <<<32, 256, 0, stream>>>(hq, cnt, pd, rs);
    k_pos <<<B_, 256, 0, stream>>>(out_n, pos_s);
    k_main<<<8192, 256, 0, stream>>>(out_n, pd, hq, rs, negm, part);
    k_final<<<1, 256, 0, stream>>>(part, pos_s, (float*)d_out);
}